// ChamferDistance2DLoss_67654324846824
// MI455X (gfx1250) — compile-verified
//
#include <hip/hip_runtime.h>
#include <math.h>

typedef float v2f __attribute__((ext_vector_type(2)));
typedef float v8f __attribute__((ext_vector_type(8)));

#define NPTS   2048
#define NBATCH 32
#define EPSF   1e-6f

// Order-preserving float->uint encoding so ds_min_u32 implements float min.
__device__ __forceinline__ unsigned f2ord(float f) {
    unsigned u = __float_as_uint(f);
    return u ^ ((unsigned)((int)u >> 31) | 0x80000000u);   // ashr + bitop3
}
__device__ __forceinline__ float ord2f(unsigned k) {
    unsigned b = (k & 0x80000000u) ? (k & 0x7fffffffu) : ~k;
    return __uint_as_float(b);
}

__global__ void ChamferZeroOut(float* out) { out[0] = 0.0f; }

__global__ __launch_bounds__(1024)
void ChamferDistance2DLoss_kernel(const float* __restrict__ pred,
                                  const float* __restrict__ target,
                                  float* __restrict__ out) {
    // B-operand staging: tb[j] = (tx, ty) for lanes 0-15 (K=0,1);
    //                    tb[NPTS+j] = (1, |t|^2) for lanes 16-31 (K=2,3).
    __shared__ float2   tb[2 * NPTS];
    __shared__ unsigned colmin[NPTS];   // running min_i dist^2(i,j), ordered-uint
    __shared__ float    redbuf[32];

    const int b   = blockIdx.x;
    const int tid = threadIdx.x;
    const float2* p2 = (const float2*)pred   + (size_t)b * NPTS;
    const float2* t2 = (const float2*)target + (size_t)b * NPTS;

    // ---- phase 0: stage B operands + colmin init ----
    for (int i = tid; i < NPTS; i += 1024) {
        float2 t = t2[i];
        tb[i]        = make_float2(t.x, t.y);
        tb[NPTS + i] = make_float2(1.0f, t.x * t.x + t.y * t.y);
        colmin[i] = 0xFFFFFFFFu;        // +inf in ordered-uint space
    }
    __syncthreads();

    const int  lane = tid & 31;
    const int  wave = tid >> 5;
    const int  lo16 = lane & 15;
    const bool hi   = lane >= 16;

    // per-lane B base: half-dependent LDS region, advances by 16 float2 per tile
    const float2* tbl = &tb[(hi ? NPTS : 0) + lo16];

    float fwd_sum = 0.0f;
    const v8f czero = {};

    // 32 waves x 4 row-tiles = 128 row-tiles of 16 rows = 2048 rows
    for (int k = 0; k < 4; ++k) {
        const int rb = (wave * 4 + k) * 16;

        // A-matrix (16x4 f32): row m = (-2px, -2py, |p_m|^2, 1)
        // lanes 0-15 supply K=0,1; lanes 16-31 supply K=2,3 (same row m = lane&15)
        float2 p   = p2[rb + lo16];
        float  pnv = p.x * p.x + p.y * p.y;
        v2f a;
        a.x = hi ? pnv  : -2.0f * p.x;
        a.y = hi ? 1.0f : -2.0f * p.y;

        float acc[8];
        #pragma unroll
        for (int v = 0; v < 8; ++v) acc[v] = INFINITY;

        #pragma unroll 4
        for (int ct = 0; ct < 128; ++ct) {
            const int cb = ct * 16;

            // one ds_load_b64: this lane's B column fragment
            float2 c = tbl[cb];
            v2f bm; bm.x = c.x; bm.y = c.y;

            // D[m][n] = |p_m|^2 + |t_n|^2 - 2 p_m.t_n  (full squared distance)
            v8f d = __builtin_amdgcn_wmma_f32_16x16x4_f32(
                false, a, false, bm, (short)0, czero, false, false);

            // INFINITY seed -> pure v_min3_num_f32 chain (no NaN-canonicalize pairs)
            float cm = INFINITY;
            #pragma unroll
            for (int v = 0; v < 8; ++v) {
                float dd = d[v];
                acc[v] = fminf(acc[v], dd);          // forward: min over column tiles
                cm     = fminf(cm, dd);              // backward: min over this row tile
            }
            // all 32 lanes atomic-min (lanes j and j+16 hit same slot)
            atomicMin(&colmin[cb + lo16], f2ord(cm));
        }

        // forward: reduce each row's min across the 16 column-lanes
        #pragma unroll
        for (int v = 0; v < 8; ++v) {
            float m = acc[v];
            m = fminf(m, __shfl_xor(m, 1, 32));
            m = fminf(m, __shfl_xor(m, 2, 32));
            m = fminf(m, __shfl_xor(m, 4, 32));
            m = fminf(m, __shfl_xor(m, 8, 32));
            acc[v] = m;
        }
        if (lo16 == 0) {  // lane 0 -> rows rb..rb+7, lane 16 -> rows rb+8..rb+15
            #pragma unroll
            for (int v = 0; v < 8; ++v)
                fwd_sum += sqrtf(fmaxf(acc[v], 0.0f) + EPSF);
        }
    }
    __syncthreads();

    // ---- backward: finish column mins ----
    float total = fwd_sum;
    for (int j = tid; j < NPTS; j += 1024) {
        float cm = ord2f(colmin[j]);
        total += sqrtf(fmaxf(cm, 0.0f) + EPSF);
    }

    // ---- block reduction + global accumulate ----
    for (int off = 16; off > 0; off >>= 1) total += __shfl_xor(total, off, 32);
    if (lane == 0) redbuf[wave] = total;
    __syncthreads();
    if (wave == 0) {
        float t = redbuf[lane];
        for (int off = 16; off > 0; off >>= 1) t += __shfl_xor(t, off, 32);
        if (lane == 0) atomicAdd(out, t * (1.0f / ((float)NBATCH * (float)NPTS)));
    }
}

extern "C" void kernel_launch(void* const* d_in, const int* in_sizes, int n_in,
                              void* d_out, int out_size, void* d_ws, size_t ws_size,
                              hipStream_t stream) {
    const float* pred   = (const float*)d_in[0];
    const float* target = (const float*)d_in[1];
    float* out = (float*)d_out;
    ChamferZeroOut<<<1, 1, 0, stream>>>(out);
    ChamferDistance2DLoss_kernel<<<NBATCH, 1024, 0, stream>>>(pred, target, out);
}